// EmbeddingBagCollection_29283087024326
// MI455X (gfx1250) — compile-verified
//
#include <hip/hip_runtime.h>
#include <cstdint>
#include <cstddef>

// ---------------- problem constants (match reference) ----------------
#define BAGS            4096
#define TABLES          4
#define DIM             64          // floats per row (256 bytes)
#define OUT_STRIDE      (TABLES * DIM)   // 256 floats per output row

// ---------------- kernel config ----------------
#define WAVES_PER_BLOCK 8
#define SLOTS           16          // async row-pairs in flight per wave (32 rows, 8KB)
#define SLOT_BYTES      512         // 2 rows * 256B
#define LDS_PER_WAVE    (SLOTS * SLOT_BYTES)             // 8 KB
#define LDS_TOTAL       (WAVES_PER_BLOCK * LDS_PER_WAVE) // 64 KB

__global__ __launch_bounds__(WAVES_PER_BLOCK * 32, 1)
void ebc_pool_kernel(const int* __restrict__ vals0, const float* __restrict__ emb0,
                     const int* __restrict__ vals1, const float* __restrict__ emb1,
                     const int* __restrict__ vals2, const float* __restrict__ emb2,
                     const int* __restrict__ vals3, const float* __restrict__ emb3,
                     const int* __restrict__ offsets,
                     float* __restrict__ out)
{
    __shared__ unsigned char smem[LDS_TOTAL];

    const int lane  = threadIdx.x & 31;
    const int wslot = threadIdx.x >> 5;

    // Table id is uniform per BLOCK -> scalar pointer select, SGPR bases.
    const int t      = (int)(blockIdx.x & 3);
    const int bgroup = (int)(blockIdx.x >> 2);              // 0 .. 511
    const int b      = bgroup * WAVES_PER_BLOCK + wslot;    // bag id, 0 .. 4095

    const int*   vals = (t == 0) ? vals0 : (t == 1) ? vals1 : (t == 2) ? vals2 : vals3;
    const float* emb  = (t == 0) ? emb0  : (t == 1) ? emb1  : (t == 2) ? emb2  : emb3;

    const int start = offsets[b];
    const int len   = offsets[b + 1] - start;

    const unsigned waveBase = (unsigned)(wslot * LDS_PER_WAVE);
    // Low 32 bits of the flat address of an LDS object == AS3 offset on amdgcn.
    const unsigned ldsBase  = (unsigned)(size_t)(const void*)smem + waveBase;

    float accx = 0.0f, accy = 0.0f;

    const int np = (len + 1) >> 1;          // number of row-pairs
    int curChunk = -1;                      // 32-index chunk currently in registers
    int chunkv   = 0;

    // Issue one async row-pair: lanes 0-15 -> row 2k, lanes 16-31 -> row 2k+1.
    auto issue = [&](int k) {
        const int chunk = k >> 4;           // 16 pairs per 32-index chunk
        if (chunk != curChunk) {
            curChunk = chunk;
            const int i = (chunk << 5) + lane;
            chunkv = (i < len) ? vals[start + i] : 0;
            // keep the (sequential) index stream ahead of us: gfx1250 global_prefetch_b8
            __builtin_prefetch(&vals[start + i + 32], 0, 1);
        }
        const int j0 = k << 1;
        const int j1 = j0 + 1;
        const int iA = __shfl(chunkv, j0 & 31, 32);
        const int iB = __shfl(chunkv, ((j1 < len) ? j1 : j0) & 31, 32);
        const unsigned idx = (unsigned)((lane < 16) ? iA : iB);

        const float* g = emb + (size_t)idx * DIM + (size_t)((lane & 15) << 2); // 16B/lane
        const unsigned dst = ldsBase + (unsigned)((k & (SLOTS - 1)) * SLOT_BYTES)
                                     + (unsigned)(lane * 16);
        asm volatile("global_load_async_to_lds_b128 %0, %1, off"
                     :: "v"(dst), "v"(g) : "memory");
    };

    // Prologue: fill the ring (16 pairs == exactly one index chunk).
    int ip = 0;
    const int pre = (np < SLOTS) ? np : SLOTS;
    for (; ip < pre; ++ip) issue(ip);

    // Steady state: wait oldest pair, accumulate from LDS, refill the freed slot.
    for (int cp = 0; cp < np; ++cp) {
        if (ip < np) {
            asm volatile("s_wait_asynccnt %0" :: "n"(SLOTS - 1) : "memory");
        } else {
            asm volatile("s_wait_asynccnt 0" ::: "memory");
        }
        const float2* slot = reinterpret_cast<const float2*>(
            smem + waveBase + (size_t)(cp & (SLOTS - 1)) * SLOT_BYTES);
        const float2 r0 = slot[lane];           // row 2*cp, cols [2*lane, 2*lane+1]
        accx += r0.x; accy += r0.y;
        if ((cp << 1) + 1 < len) {              // uniform branch per wave
            const float2 r1 = slot[32 + lane];  // row 2*cp+1
            accx += r1.x; accy += r1.y;
        }
        if (ip < np) {
            // our DS reads of this slot must retire before the DMA may rewrite it
            asm volatile("s_wait_dscnt 0" ::: "memory");
            issue(ip++);
        }
    }

    float scale = 1.0f;
    if (t >= 2) scale = (len > 0) ? (1.0f / (float)len) : 0.0f;  // mean pooling (scalar branch)

    float2 o;
    o.x = accx * scale;
    o.y = accy * scale;
    float2* op = reinterpret_cast<float2*>(out + (size_t)b * OUT_STRIDE + (size_t)t * DIM);
    op[lane] = o;
}

extern "C" void kernel_launch(void* const* d_in, const int* in_sizes, int n_in,
                              void* d_out, int out_size, void* d_ws, size_t ws_size,
                              hipStream_t stream) {
    (void)in_sizes; (void)n_in; (void)out_size; (void)d_ws; (void)ws_size;
    // setup_inputs() dict order: values0, emb0, values1, emb1, values2, emb2,
    //                            values3, emb3, offsets
    const int*   v0  = (const int*)  d_in[0];
    const float* e0  = (const float*)d_in[1];
    const int*   v1  = (const int*)  d_in[2];
    const float* e1  = (const float*)d_in[3];
    const int*   v2  = (const int*)  d_in[4];
    const float* e2  = (const float*)d_in[5];
    const int*   v3  = (const int*)  d_in[6];
    const float* e3  = (const float*)d_in[7];
    const int*   off = (const int*)  d_in[8];
    float* out = (float*)d_out;

    const int tasks  = BAGS * TABLES;                 // 16384 waves
    const int blocks = tasks / WAVES_PER_BLOCK;       // 2048 (t = blockIdx & 3)
    hipLaunchKernelGGL(ebc_pool_kernel, dim3(blocks), dim3(WAVES_PER_BLOCK * 32), 0, stream,
                       v0, e0, v1, e1, v2, e2, v3, e3, off, out);
}